// CriticNetwork_6365141533089
// MI455X (gfx1250) — compile-verified
//
#include <hip/hip_runtime.h>

typedef __attribute__((ext_vector_type(2))) float v2f;
typedef __attribute__((ext_vector_type(8))) float v8f;

#define NB 4
#define N0 50000
#define NODES_TOTAL 200000   // B * N
#define FN 64
#define HD 16
#define NC 1000
#define FC 32

// ---------------- degree count over dst (edges only touch [0, N0)) ----------
__global__ void k_deg_count(const int* __restrict__ dst, unsigned* __restrict__ cnt, int E) {
    int e = blockIdx.x * blockDim.x + threadIdx.x;
    if (e < E) atomicAdd(&cnt[dst[e]], 1u);
}

// deg[i] = 4*cnt[i] + 1 for i < N0 (edges duplicated 4x by batch broadcast), else 1 (self loop only)
__global__ void k_dinv(const unsigned* __restrict__ cnt, float* __restrict__ dinv, int n) {
    int i = blockIdx.x * blockDim.x + threadIdx.x;
    if (i >= n) return;
    float deg = (i < N0) ? (float)(4u * cnt[i] + 1u) : 1.0f;
    dinv[i] = rsqrtf(deg);
}

// ---------------- f32 WMMA GEMM: Y[ntiles*16, 16] = X[ntiles*16, KD] @ W[KD, 16] ----------
// One wave per 16x16 output tile, K stepped by 4 via V_WMMA_F32_16X16X4_F32.
template<int KD>
__global__ void k_gemm_wmma(const float* __restrict__ X, const float* __restrict__ W,
                            float* __restrict__ Y, int ntiles) {
    int wave = (int)(((unsigned)blockIdx.x * blockDim.x + threadIdx.x) >> 5);
    if (wave >= ntiles) return;                 // uniform per wave -> EXEC stays all-1s
    int lane = threadIdx.x & 31;
    int half = lane >> 4;                       // 0: K=k0,k0+1   1: K=k0+2,k0+3
    int m    = lane & 15;                       // row (A) / col (B,C)
    int row0 = wave << 4;
    const float* xrow = X + (size_t)(row0 + m) * KD;
    v8f acc = {};
#pragma unroll
    for (int k0 = 0; k0 < KD; k0 += 4) {
        int k = k0 + 2 * half;
        v2f a, b;
        a.x = xrow[k];
        a.y = xrow[k + 1];
        b.x = W[k * HD + m];
        b.y = W[(k + 1) * HD + m];
        acc = __builtin_amdgcn_wmma_f32_16x16x4_f32(false, a, false, b, (short)0, acc,
                                                    false, false);
    }
#pragma unroll
    for (int v = 0; v < 8; ++v) {               // C/D layout: VGPR v -> M = v + 8*half, N = m
        Y[(size_t)(row0 + v + 8 * half) * HD + m] = acc[v];
    }
}

// ---------------- edge scatter-add: agg[d] += dinv[s] * Y[s], 16 lanes per edge -------------
__global__ void k_edge_agg(const int* __restrict__ src, const int* __restrict__ dst,
                           const float* __restrict__ dinv, const float* __restrict__ Y,
                           float* __restrict__ agg, int E) {
    long long t = (long long)blockIdx.x * blockDim.x + threadIdx.x;
    int e = (int)(t >> 4);
    if (e >= E) return;
    int j = (int)(t & 15);
    int s = src[e];
    int d = dst[e];
    atomicAdd(&agg[(size_t)d * HD + j], dinv[s] * Y[(size_t)s * HD + j]);
}

// H = relu(dinv^2 * Y + (i<N0 ? 4*dinv*agg : 0) + bias)
__global__ void k_combine(const float* __restrict__ Y, const float* __restrict__ agg,
                          const float* __restrict__ dinv, const float* __restrict__ bias,
                          float* __restrict__ H) {
    long long t = (long long)blockIdx.x * blockDim.x + threadIdx.x;
    int i = (int)(t >> 4);
    if (i >= NODES_TOTAL) return;
    int j = (int)(t & 15);
    float di = dinv[i];
    float v = di * di * Y[t] + bias[j];
    if (i < N0) v += 4.0f * di * agg[t];
    H[t] = fmaxf(v, 0.0f);
}

// conv2 epilogue fused with node_fc head and per-batch mean reduction
__global__ void k_node_out(const float* __restrict__ Y2, const float* __restrict__ agg,
                           const float* __restrict__ dinv, const float* __restrict__ b2,
                           const float* __restrict__ fcW, const float* __restrict__ fcB,
                           float* __restrict__ node_sum) {
    __shared__ float bs[NB];
    if (threadIdx.x < NB) bs[threadIdx.x] = 0.0f;
    __syncthreads();
    int i = blockIdx.x * blockDim.x + threadIdx.x;
    if (i < NODES_TOTAL) {
        float di = dinv[i];
        float sc = di * di;
        bool has = (i < N0);
        float s4 = has ? 4.0f * di : 0.0f;
        float acc = fcB[0];
#pragma unroll
        for (int j = 0; j < HD; ++j) {
            float a = has ? agg[(size_t)i * HD + j] : 0.0f;
            float v = sc * Y2[(size_t)i * HD + j] + s4 * a + b2[j];
            acc += fmaxf(v, 0.0f) * fcW[j];
        }
        atomicAdd(&bs[i / N0], acc);
    }
    __syncthreads();
    if (threadIdx.x < NB) atomicAdd(&node_sum[threadIdx.x], bs[threadIdx.x]);
}

// column branch: relu(colf @ cW1 + cB1) @ cW2 + cB2, per-batch sum
__global__ void k_col(const float* __restrict__ colf, const float* __restrict__ W1,
                      const float* __restrict__ B1, const float* __restrict__ W2,
                      const float* __restrict__ B2, float* __restrict__ col_sum) {
    __shared__ float bs[NB];
    if (threadIdx.x < NB) bs[threadIdx.x] = 0.0f;
    __syncthreads();
    int t = blockIdx.x * blockDim.x + threadIdx.x;
    if (t < NB * NC) {
        const float* xr = colf + (size_t)t * FC;
        float out = B2[0];
#pragma unroll
        for (int j = 0; j < HD; ++j) {
            float h = B1[j];
#pragma unroll
            for (int k = 0; k < FC; ++k) h += xr[k] * W1[k * HD + j];
            out += fmaxf(h, 0.0f) * W2[j];
        }
        atomicAdd(&bs[t / NC], out);
    }
    __syncthreads();
    if (threadIdx.x < NB) atomicAdd(&col_sum[threadIdx.x], bs[threadIdx.x]);
}

// final tiny MLP: [node_avg, col_avg] @ fc_W + fc_b -> relu -> @ out_W + out_b
__global__ void k_final(const float* __restrict__ node_sum, const float* __restrict__ col_sum,
                        const float* __restrict__ fcW, const float* __restrict__ fcB,
                        const float* __restrict__ outW, const float* __restrict__ outB,
                        float* __restrict__ out) {
    int b = threadIdx.x;
    if (b >= NB) return;
    float na = node_sum[b] * (1.0f / (float)N0);
    float ca = col_sum[b] * (1.0f / (float)NC);
    float o = outB[0];
#pragma unroll
    for (int j = 0; j < HD; ++j) {
        float z = na * fcW[j] + ca * fcW[HD + j] + fcB[j];
        o += fmaxf(z, 0.0f) * outW[j];
    }
    out[b] = o;
}

extern "C" void kernel_launch(void* const* d_in, const int* in_sizes, int n_in,
                              void* d_out, int out_size, void* d_ws, size_t ws_size,
                              hipStream_t stream) {
    (void)n_in; (void)out_size; (void)ws_size;
    const float* node_f = (const float*)d_in[0];
    const float* col_f  = (const float*)d_in[1];
    const int*   eidx   = (const int*)d_in[2];
    const float* W1     = (const float*)d_in[3];
    const float* b1     = (const float*)d_in[4];
    const float* W2     = (const float*)d_in[5];
    const float* b2     = (const float*)d_in[6];
    const float* nfcW   = (const float*)d_in[7];
    const float* nfcB   = (const float*)d_in[8];
    const float* cW1    = (const float*)d_in[9];
    const float* cB1    = (const float*)d_in[10];
    const float* cW2    = (const float*)d_in[11];
    const float* cB2    = (const float*)d_in[12];
    const float* fcW    = (const float*)d_in[13];
    const float* fcB    = (const float*)d_in[14];
    const float* outW   = (const float*)d_in[15];
    const float* outB   = (const float*)d_in[16];

    const int E = in_sizes[2] / 2;
    const int* src = eidx;
    const int* dst = eidx + E;

    // workspace layout (256B aligned chunks), ~29.8 MB total
    char* ws = (char*)d_ws;
    size_t off = 0;
    auto take = [&](size_t bytes) -> char* {
        char* p = ws + off;
        off = (off + bytes + 255) & ~(size_t)255;
        return p;
    };
    unsigned* cnt   = (unsigned*)take((size_t)N0 * 4);
    float* dinv     = (float*)take((size_t)NODES_TOTAL * 4);
    float* Y        = (float*)take((size_t)NODES_TOTAL * HD * 4);  // y1 then reused as y2
    float* Hbuf     = (float*)take((size_t)NODES_TOTAL * HD * 4);  // relu(conv1)
    float* agg      = (float*)take((size_t)N0 * HD * 4);           // reused for conv2
    float* sums     = (float*)take(2 * NB * 4);
    float* node_sum = sums;
    float* col_sum  = sums + NB;

    hipMemsetAsync(cnt, 0, (size_t)N0 * 4, stream);
    hipMemsetAsync(sums, 0, 2 * NB * 4, stream);

    // degrees -> dinv
    k_deg_count<<<(E + 255) / 256, 256, 0, stream>>>(dst, cnt, E);
    k_dinv<<<(NODES_TOTAL + 255) / 256, 256, 0, stream>>>(cnt, dinv, NODES_TOTAL);

    const int ntiles = NODES_TOTAL / 16;                 // 12500 16-row tiles
    const int gemm_blocks = (ntiles + 7) / 8;            // 8 waves / 256-thread block
    const long long etot = (long long)E * HD;
    const unsigned eblocks = (unsigned)((etot + 255) / 256);
    const long long ctot = (long long)NODES_TOTAL * HD;
    const unsigned cblocks = (unsigned)((ctot + 255) / 256);

    // ---- conv1 ----
    k_gemm_wmma<FN><<<gemm_blocks, 256, 0, stream>>>(node_f, W1, Y, ntiles);
    hipMemsetAsync(agg, 0, (size_t)N0 * HD * 4, stream);
    k_edge_agg<<<eblocks, 256, 0, stream>>>(src, dst, dinv, Y, agg, E);
    k_combine<<<cblocks, 256, 0, stream>>>(Y, agg, dinv, b1, Hbuf);

    // ---- conv2 (reuse Y for y2, agg for agg2) ----
    k_gemm_wmma<HD><<<gemm_blocks, 256, 0, stream>>>(Hbuf, W2, Y, ntiles);
    hipMemsetAsync(agg, 0, (size_t)N0 * HD * 4, stream);
    k_edge_agg<<<eblocks, 256, 0, stream>>>(src, dst, dinv, Y, agg, E);

    // ---- heads + reductions ----
    k_node_out<<<(NODES_TOTAL + 255) / 256, 256, 0, stream>>>(Y, agg, dinv, b2, nfcW, nfcB,
                                                              node_sum);
    k_col<<<(NB * NC + 255) / 256, 256, 0, stream>>>(col_f, cW1, cB1, cW2, cB2, col_sum);
    k_final<<<1, 32, 0, stream>>>(node_sum, col_sum, fcW, fcB, outW, outB, (float*)d_out);
}